// BS2_36618891166097
// MI455X (gfx1250) — compile-verified
//
#include <hip/hip_runtime.h>

// B-spline (ncp=32, degree=3) rational evaluation, fp32 end-to-end.
// dp[b,t,:] = (N[t,:] @ (cp*w)) / (N[t,:] @ w + eps)
// Matmul via chained V_WMMA_F32_16X16X4_F32 (K=32 = 8 x K=4 steps),
// two interleaved accumulator chains, A-fragments preloaded to registers.

#define NCP   32
#define DEG   3
#define EPSF  1e-7f
#define NT    192          // t-values per batch
#define NWAVE 6            // 192 threads / wave32
#define NSTR  33           // padded LDS row stride for basis rows

typedef __attribute__((ext_vector_type(2))) float v2f;
typedef __attribute__((ext_vector_type(8))) float v8f;

// knots: [0,0,0,0, 1/29 .. 28/29, 1,1,1,1]  (36 values, index 0..35)
// knot(i) = clamp((i-3)/29, 0, 1)  -- constant-folds under full unroll.
__device__ __forceinline__ float knotf(int i) {
    float v = (float)(i - DEG) * (1.0f / 29.0f);
    v = v < 0.0f ? 0.0f : v;
    v = v > 1.0f ? 1.0f : v;
    return v;
}

__global__ __launch_bounds__(NT)
void bspline_wmma_kernel(const float* __restrict__ cp,   // [B,32,3]
                         const float* __restrict__ w,    // [B,32,1]
                         const float* __restrict__ ub,   // [B,192]
                         float* __restrict__ out,        // [B,192,3]
                         int B)
{
    __shared__ float sCW[NCP * 4];              // rows k: {cp.x*w, cp.y*w, cp.z*w, w}
    __shared__ float sN[NWAVE][32 * NSTR];      // per-wave basis rows (32 t x 32 k)
    __shared__ float sD[NWAVE][32 * 4];         // per-wave results (32 t x {x,y,z,nw})

    const int b     = blockIdx.x;
    const int tid   = threadIdx.x;
    const int wv    = tid >> 5;                 // wave id 0..5
    const int lane  = tid & 31;
    const int col   = lane & 15;                // WMMA lane column / A-row selector
    const int half  = lane >> 4;                // WMMA lane half
    const int tbase = wv * 32;

    // ---- stage B-matrix data: cp*w and w (wave 0, one row per lane) ----
    if (tid < NCP) {
        const int k = tid;
        const float wt = w[(size_t)b * NCP + k];
        const float* c = cp + ((size_t)b * NCP + k) * 3;
        sCW[k * 4 + 0] = c[0] * wt;
        sCW[k * 4 + 1] = c[1] * wt;
        sCW[k * 4 + 2] = c[2] * wt;
        sCW[k * 4 + 3] = wt;
    }

    // ---- per-lane Cox-de Boor DP (fp32, all knot reciprocals constant-folded) ----
    const float t = ub[(size_t)b * NT + tbase + lane];

    float Np[NCP + DEG];                        // 35 degree-0 indicators
#pragma unroll
    for (int i = 0; i < NCP + DEG; ++i)
        Np[i] = (knotf(i) <= t && t < knotf(i + 1)) ? 1.0f : 0.0f;

    auto step = [&](int p, int i) {
        const float invL = 1.0f / (knotf(i + p)     - knotf(i)     + EPSF);
        const float invR = 1.0f / (knotf(i + p + 1) - knotf(i + 1) + EPSF);
        Np[i] = (t - knotf(i)) * invL * Np[i]
              + (knotf(i + p + 1) - t) * invR * Np[i + 1];
    };
#pragma unroll
    for (int i = 0; i < 34; ++i) step(1, i);
#pragma unroll
    for (int i = 0; i < 33; ++i) step(2, i);
#pragma unroll
    for (int i = 0; i < 32; ++i) step(3, i);

    float* nrow = &sN[wv][lane * NSTR];
#pragma unroll
    for (int i = 0; i < NCP; ++i) nrow[i] = Np[i];

    __syncthreads();

    // ---- build 8 B-fragments (K-chunks of 4): v0 = B[k0+2*half][col], v1 = row+1 ----
    const int cc = (col < 4) ? col : 0;         // clamp address, mask value
    v2f bf[8];
#pragma unroll
    for (int kc = 0; kc < 8; ++kc) {
        const int k0 = kc * 4 + half * 2;
        float b0 = sCW[(k0 + 0) * 4 + cc];
        float b1 = sCW[(k0 + 1) * 4 + cc];
        bf[kc].x = (col < 4) ? b0 : 0.0f;
        bf[kc].y = (col < 4) ? b1 : 0.0f;
    }

    // ---- preload ALL A-fragments (both 16-row tiles) into registers ----
    const float* nbase = sN[wv];
    v2f af[2][8];
#pragma unroll
    for (int tile = 0; tile < 2; ++tile) {
        const float* arow = nbase + (tile * 16 + col) * NSTR + half * 2;
#pragma unroll
        for (int kc = 0; kc < 8; ++kc) {
            af[tile][kc].x = arow[kc * 4 + 0];  // A vgpr0: K = 4kc + 2*half
            af[tile][kc].y = arow[kc * 4 + 1];  // A vgpr1: K = 4kc + 2*half + 1
        }
    }

    // ---- two interleaved fp32 WMMA accumulation chains ----
    v8f acc0 = {0.f, 0.f, 0.f, 0.f, 0.f, 0.f, 0.f, 0.f};
    v8f acc1 = {0.f, 0.f, 0.f, 0.f, 0.f, 0.f, 0.f, 0.f};
#pragma unroll
    for (int kc = 0; kc < 8; ++kc) {
        acc0 = __builtin_amdgcn_wmma_f32_16x16x4_f32(
            false, af[0][kc], false, bf[kc], (short)0, acc0, false, false);
        acc1 = __builtin_amdgcn_wmma_f32_16x16x4_f32(
            false, af[1][kc], false, bf[kc], (short)0, acc1, false, false);
    }

    // D layout: lane col = N, vgpr j holds row M = j + 8*half. Keep cols 0..3.
    if (col < 4) {
#pragma unroll
        for (int j = 0; j < 8; ++j) {
            sD[wv][(half * 8 + j) * 4 + col]        = acc0[j];
            sD[wv][(16 + half * 8 + j) * 4 + col]   = acc1[j];
        }
    }

    __syncthreads();

    // ---- normalize & write: lane L owns t-row L of this wave ----
    const float dx = sD[wv][lane * 4 + 0];
    const float dy = sD[wv][lane * 4 + 1];
    const float dz = sD[wv][lane * 4 + 2];
    const float nw = sD[wv][lane * 4 + 3];
    const float inv = 1.0f / (nw + EPSF);

    const size_t o = ((size_t)b * NT + tbase + lane) * 3;
    out[o + 0] = dx * inv;
    out[o + 1] = dy * inv;
    out[o + 2] = dz * inv;
}

extern "C" void kernel_launch(void* const* d_in, const int* in_sizes, int n_in,
                              void* d_out, int out_size, void* d_ws, size_t ws_size,
                              hipStream_t stream) {
    const float* cp = (const float*)d_in[0];
    const float* w  = (const float*)d_in[1];
    const float* ub = (const float*)d_in[2];
    float* out = (float*)d_out;

    const int B = in_sizes[2] / NT;             // 8192 in the reference setup
    dim3 grid((unsigned)B), block(NT);
    hipLaunchKernelGGL(bspline_wmma_kernel, grid, block, 0, stream,
                       cp, w, ub, out, B);
}